// GMM_5798205850046
// MI455X (gfx1250) — compile-verified
//
#include <hip/hip_runtime.h>
#include <math.h>

// GMM log-prob: K=32 comps, D=64, N=65536.
// weighted[n,k] = sum_e w_e[k] * f_e(x_n), f_e = {x_i*x_j (tri-blocked), x_i, 1}
// computed as fp32 WMMA GEMM [16 x 2244] x [2244 x 32]; out[n] = lse_k weighted.
// Each wave register-blocks RB=4 row-tiles (64 points) to cut L2 B-pack traffic 4x.

#define NGROUPS 561   // 544 pair groups (sum_j (j/4+1)) + 16 linear + 1 const
#define GMM_K   32
#define LOG2PI  1.8378770664093453f
#define XS      68    // LDS row stride in floats (16B-aligned rows, conflict-free)
#define RB      4     // row tiles (of 16 points) per wave
#define WPB     3     // waves per block (LDS: 3*64*68*4 = 52.2 KB)

typedef __attribute__((ext_vector_type(2))) float v2f;
typedef __attribute__((ext_vector_type(8))) float v8f;

// ---------------- Phase 1: build per-component coefficient pack ----------------
__global__ __launch_bounds__(64) void gmm_prep(const float* __restrict__ loc,
                                               const float* __restrict__ cov,
                                               const float* __restrict__ pi,
                                               float* __restrict__ bpack) {
  __shared__ float A[64][65];   // cov -> L -> P (lower)
  __shared__ float M[64][65];   // L^{-1}
  __shared__ float mu[64];
  __shared__ float bv[64];      // P * mu
  __shared__ float sc[3];       // lse(pi), half_logdet, const-term
  const int k = blockIdx.x;
  const int t = threadIdx.x;

  for (int i = 0; i < 64; ++i) A[t][i] = cov[((size_t)k * 64 + t) * 64 + i];
  mu[t] = loc[k * 64 + t];
  if (t == 0) {
    float m = pi[0];
    for (int i = 1; i < GMM_K; ++i) m = fmaxf(m, pi[i]);
    float s = 0.f;
    for (int i = 0; i < GMM_K; ++i) s += expf(pi[i] - m);
    sc[0] = m + logf(s);
  }
  __syncthreads();

  // Cholesky (in-place, lower), thread t owns row t
  for (int j = 0; j < 64; ++j) {
    if (t == j) A[j][j] = sqrtf(A[j][j]);
    __syncthreads();
    if (t > j) A[t][j] /= A[j][j];
    __syncthreads();
    if (t > j) {
      float l = A[t][j];
      for (int c = j + 1; c <= t; ++c) A[t][c] -= l * A[c][j];
    }
    __syncthreads();
  }
  if (t == 0) {
    float hs = 0.f;
    for (int i = 0; i < 64; ++i) hs += logf(A[i][i]);
    sc[1] = hs;
  }

  // M = L^{-1}: thread t solves column t by forward substitution
  {
    const int c = t;
    for (int i = 0; i < c; ++i) M[i][c] = 0.f;
    M[c][c] = 1.f / A[c][c];
    for (int i = c + 1; i < 64; ++i) {
      float s = 0.f;
      for (int p = c; p < i; ++p) s += A[i][p] * M[p][c];
      M[i][c] = -s / A[i][i];
    }
  }
  __syncthreads();

  // P = M^T M, lower triangle into A: thread j computes A[j][i], i<=j
  {
    const int j = t;
    for (int i = 0; i <= j; ++i) {
      float s = 0.f;
      for (int p = j; p < 64; ++p) s += M[p][i] * M[p][j];
      A[j][i] = s;
    }
  }
  __syncthreads();

  // bv = P * mu
  {
    const int a = t;
    float s = 0.f;
    for (int b = 0; b < 64; ++b) {
      float p = (b <= a) ? A[a][b] : A[b][a];
      s += p * mu[b];
    }
    bv[a] = s;
  }
  __syncthreads();
  if (t == 0) {
    float c2 = 0.f;
    for (int i = 0; i < 64; ++i) c2 += mu[i] * bv[i];
    sc[2] = -0.5f * c2 + pi[k] - sc[0] - sc[1] - 0.5f * 64.f * LOG2PI;
  }
  __syncthreads();

  // Emit B-pack in WMMA B-fragment layout:
  // fragment (ctile,g): 64 floats = [h:2][n:16][q':2]; entry (q=2h+q', comp n)
  const int ctile = k >> 4;
  const int n = k & 15;
  float* base = bpack + (size_t)ctile * NGROUPS * 64;
  int g = 0;
  for (int j = 0; j < 64; ++j) {                 // pair groups: f = x_i * x_j
    const int nc = (j >> 2) + 1;
    for (int ci = 0; ci < nc; ++ci, ++g) {
      if ((g & 63) == t) {
        float* dst = base + (size_t)g * 64 + n * 2;
        for (int q = 0; q < 4; ++q) {
          const int i = 4 * ci + q;
          float w = 0.f;
          if (i < j)       w = -A[j][i];         // off-diag pair coeff
          else if (i == j) w = -0.5f * A[j][j];  // diagonal coeff
          dst[(q >> 1) * 32 + (q & 1)] = w;
        }
      }
    }
  }
  for (int ci = 0; ci < 16; ++ci, ++g) {         // linear groups: f = x_i
    if ((g & 63) == t) {
      float* dst = base + (size_t)g * 64 + n * 2;
      for (int q = 0; q < 4; ++q)
        dst[(q >> 1) * 32 + (q & 1)] = bv[4 * ci + q];
    }
  }
  if ((g & 63) == t) {                           // const group (g == 560)
    float* dst = base + (size_t)g * 64 + n * 2;
    for (int q = 0; q < 4; ++q)
      dst[(q >> 1) * 32 + (q & 1)] = (q == 0) ? sc[2] : 0.f;
  }
}

// ---------------- Phase 2: WMMA GEMM + logsumexp, RB=4 row blocking ----------------
// Block = 96 threads = 3 waves; each wave owns 4 consecutive 16-point tiles.
__global__ __launch_bounds__(32 * WPB) void gmm_main(const float* __restrict__ X,
                                                     const float* __restrict__ bpack,
                                                     float* __restrict__ out,
                                                     int ntiles) {
  __shared__ float xs[WPB][16 * RB][XS];
  const int tid = threadIdx.x;
  const int w = tid >> 5;
  const int lane = tid & 31;
  const int h = lane >> 4;     // wave half -> K elements 2h, 2h+1 of each group
  const int r = lane & 15;     // A fragment row (point within tile), C fragment column (component)
  const int t0 = (blockIdx.x * WPB + w) * RB;   // first tile of this wave
  int nvalid = ntiles - t0;                     // # valid tiles for this wave
  if (nvalid > RB) nvalid = RB;

  if (nvalid > 0) {  // stage 16*nvalid points (coalesced) into padded LDS rows
    const float* xp = X + (size_t)t0 * 16 * 64;
    const int umax = nvalid * 8;                // each u covers 128 floats (2 rows)
    for (int u = 0; u < umax; ++u) {
      const int flat = u * 128 + lane * 4;
      float4 v = *(const float4*)(xp + flat);
      *(float4*)&xs[w][flat >> 6][flat & 63] = v;
    }
  }
  __syncthreads();
  if (nvalid <= 0) return;

  const float* xr[RB];
#pragma unroll
  for (int rt = 0; rt < RB; ++rt) xr[rt] = &xs[w][16 * rt + r][0];
  const v2f* __restrict__ bp0 = (const v2f*)bpack + (h * 16 + r);  // comp tile 0
  const v2f* __restrict__ bp1 = bp0 + (size_t)NGROUPS * 32;        // comp tile 1

  v8f acc[RB][2];
#pragma unroll
  for (int rt = 0; rt < RB; ++rt) { acc[rt][0] = (v8f){}; acc[rt][1] = (v8f){}; }

  int g = 0;
  for (int j = 0; j < 64; ++j) {                 // pair groups: f = x_i * x_j
    float tj[RB];
#pragma unroll
    for (int rt = 0; rt < RB; ++rt) tj[rt] = xr[rt][j];
    const int nc = (j >> 2) + 1;
    for (int ci = 0; ci < nc; ++ci, ++g) {
      v2f b0 = bp0[(size_t)g * 32];
      v2f b1 = bp1[(size_t)g * 32];
      v2f a[RB];
#pragma unroll
      for (int rt = 0; rt < RB; ++rt)
        a[rt] = (*(const v2f*)(xr[rt] + 4 * ci + 2 * h)) * tj[rt];
#pragma unroll
      for (int rt = 0; rt < RB; ++rt) {
        acc[rt][0] = __builtin_amdgcn_wmma_f32_16x16x4_f32(false, a[rt], false, b0, (short)0, acc[rt][0], false, false);
        acc[rt][1] = __builtin_amdgcn_wmma_f32_16x16x4_f32(false, a[rt], false, b1, (short)0, acc[rt][1], false, false);
      }
    }
  }
  for (int ci = 0; ci < 16; ++ci, ++g) {         // linear groups: f = x_i
    v2f b0 = bp0[(size_t)g * 32];
    v2f b1 = bp1[(size_t)g * 32];
    v2f a[RB];
#pragma unroll
    for (int rt = 0; rt < RB; ++rt)
      a[rt] = *(const v2f*)(xr[rt] + 4 * ci + 2 * h);
#pragma unroll
    for (int rt = 0; rt < RB; ++rt) {
      acc[rt][0] = __builtin_amdgcn_wmma_f32_16x16x4_f32(false, a[rt], false, b0, (short)0, acc[rt][0], false, false);
      acc[rt][1] = __builtin_amdgcn_wmma_f32_16x16x4_f32(false, a[rt], false, b1, (short)0, acc[rt][1], false, false);
    }
  }
  {                                              // const group: f = 1
    v2f b0 = bp0[(size_t)g * 32];
    v2f b1 = bp1[(size_t)g * 32];
    v2f a; a.x = (h == 0) ? 1.f : 0.f; a.y = 0.f;
#pragma unroll
    for (int rt = 0; rt < RB; ++rt) {
      acc[rt][0] = __builtin_amdgcn_wmma_f32_16x16x4_f32(false, a, false, b0, (short)0, acc[rt][0], false, false);
      acc[rt][1] = __builtin_amdgcn_wmma_f32_16x16x4_f32(false, a, false, b1, (short)0, acc[rt][1], false, false);
    }
  }

  // logsumexp over 32 comps: 16 lane-columns x 2 accumulators, per C-row v
#pragma unroll
  for (int rt = 0; rt < RB; ++rt) {
    float res[8];
#pragma unroll
    for (int v = 0; v < 8; ++v) {
      float a0 = acc[rt][0][v], a1 = acc[rt][1][v];
      float mx = fmaxf(a0, a1);
      mx = fmaxf(mx, __shfl_xor(mx, 1, 32));
      mx = fmaxf(mx, __shfl_xor(mx, 2, 32));
      mx = fmaxf(mx, __shfl_xor(mx, 4, 32));
      mx = fmaxf(mx, __shfl_xor(mx, 8, 32));
      float e = expf(a0 - mx) + expf(a1 - mx);
      e += __shfl_xor(e, 1, 32);
      e += __shfl_xor(e, 2, 32);
      e += __shfl_xor(e, 4, 32);
      e += __shfl_xor(e, 8, 32);
      res[v] = mx + logf(e);
    }
    float val = res[0];
#pragma unroll
    for (int v = 1; v < 8; ++v) val = (r == v) ? res[v] : val;
    if (rt < nvalid && r < 8)
      out[(size_t)(t0 + rt) * 16 + 8 * h + r] = val;   // rows 8h+r, coalesced
  }
}

// ---------------- Launch ----------------
extern "C" void kernel_launch(void* const* d_in, const int* in_sizes, int n_in,
                              void* d_out, int out_size, void* d_ws, size_t ws_size,
                              hipStream_t stream) {
  const float* X   = (const float*)d_in[0];
  const float* loc = (const float*)d_in[1];
  const float* cov = (const float*)d_in[2];
  const float* pi  = (const float*)d_in[3];
  float* out = (float*)d_out;
  float* bpack = (float*)d_ws;    // 2 * NGROUPS * 64 floats = ~287 KB

  const int N = in_sizes[0] / 64;
  const int ntiles = N / 16;

  gmm_prep<<<GMM_K, 64, 0, stream>>>(loc, cov, pi, bpack);
  const int tilesPerBlock = WPB * RB;
  const int nblk = (ntiles + tilesPerBlock - 1) / tilesPerBlock;
  gmm_main<<<nblk, 32 * WPB, 0, stream>>>(X, bpack, out, ntiles);
}